// SpuriousTranslationLayerV9_45165876084913
// MI455X (gfx1250) — compile-verified
//
#include <hip/hip_runtime.h>
#include <hip/hip_bf16.h>

typedef __attribute__((ext_vector_type(2))) float v2f;
typedef __attribute__((ext_vector_type(8))) float v8f;

#define BATCH   2048
#define NSEEN   32768
#define HIDDEN  768
#define TOPK    64
#define NTHR    256
#define VSTRIDE 770   // padded LDS row stride in floats (768 would alias banks 16-way)

// ---------------------------------------------------------------------------
// Kernel 1: exact per-row top-64 via 3-pass radix select, row staged in LDS.
// One 256-thread workgroup per row; dynamic LDS ~140KB (row 128KB + hist 8KB).
// ---------------------------------------------------------------------------
__global__ __launch_bounds__(NTHR)
void topk_kernel(const float* __restrict__ s,
                 int* __restrict__ out_idx, float* __restrict__ out_val)
{
    extern __shared__ unsigned char smem[];
    float*    srow = (float*)smem;                                   // 32768 f32
    unsigned* hist = (unsigned*)(smem + (size_t)NSEEN * 4);          // 2048
    unsigned* bsum = (unsigned*)((unsigned char*)hist + 2048 * 4);   // 256
    int*      lidx = (int*)((unsigned char*)bsum + 256 * 4);         // 64
    float*    lval = (float*)((unsigned char*)lidx + 64 * 4);        // 64
    unsigned* vars = (unsigned*)((unsigned char*)lval + 64 * 4);     // [0]=k [1]=selbin [2]=cnt_gt [3]=cnt_eq

    const int tid = threadIdx.x;
    const int row = blockIdx.x;

    // Stage row into LDS (coalesced float4)
    const float4* g  = (const float4*)(s + (size_t)row * NSEEN);
    float4*       l4 = (float4*)srow;
    #pragma unroll
    for (int i = 0; i < NSEEN / 4 / NTHR; ++i)
        l4[i * NTHR + tid] = g[i * NTHR + tid];
    if (tid == 0) { vars[0] = TOPK; vars[2] = 0u; vars[3] = 0u; }
    __syncthreads();

    unsigned sel[3];
    for (int pass = 0; pass < 3; ++pass) {
        #pragma unroll
        for (int j = 0; j < 8; ++j) hist[tid * 8 + j] = 0u;
        __syncthreads();

        // Histogram candidates (strided LDS access -> conflict-free)
        for (int j = 0; j < NSEEN / NTHR; ++j) {
            unsigned u = __float_as_uint(srow[j * NTHR + tid]);
            unsigned bin; bool cand;
            if (pass == 0)      { cand = true;                                    bin = u >> 20; }
            else if (pass == 1) { cand = ((u >> 20) == sel[0]);                   bin = (u >> 9) & 0x7FFu; }
            else                { cand = ((u >> 9) == ((sel[0] << 11) | sel[1])); bin = u & 0x1FFu; }
            if (cand) atomicAdd(&hist[bin], 1u);
        }
        __syncthreads();

        // Find bin containing the k-th largest (suffix counts from the top)
        unsigned ssum = 0;
        #pragma unroll
        for (int j = 0; j < 8; ++j) ssum += hist[tid * 8 + j];
        bsum[tid] = ssum;
        unsigned kcur = vars[0];
        __syncthreads();
        unsigned sf = 0;
        for (int t2 = tid + 1; t2 < NTHR; ++t2) sf += bsum[t2];
        if (sf < kcur && sf + ssum >= kcur) {   // unique owner thread
            unsigned cum = sf;
            #pragma unroll
            for (int j = 7; j >= 0; --j) {
                unsigned c = hist[tid * 8 + j];
                if (cum + c >= kcur) { vars[1] = (unsigned)(tid * 8 + j); vars[0] = kcur - cum; break; }
                cum += c;
            }
        }
        __syncthreads();
        sel[pass] = vars[1];
    }

    const unsigned T  = (sel[0] << 20) | (sel[1] << 9) | sel[2];  // exact 64th value bits
    const unsigned kT = vars[0];                                  // # needed among == T

    for (int j = 0; j < NSEEN / NTHR; ++j) {
        int e = j * NTHR + tid;
        unsigned u = __float_as_uint(srow[e]);
        if (u > T) {
            unsigned pos = atomicAdd(&vars[2], 1u);   // count(>T) < 64 by construction
            lidx[pos] = e; lval[pos] = __uint_as_float(u);
        }
    }
    __syncthreads();
    unsigned base = vars[2];
    for (int j = 0; j < NSEEN / NTHR; ++j) {
        int e = j * NTHR + tid;
        unsigned u = __float_as_uint(srow[e]);
        if (u == T) {
            unsigned pos = atomicAdd(&vars[3], 1u);
            if (pos < kT) { lidx[base + pos] = e; lval[base + pos] = __uint_as_float(u); }
        }
    }
    __syncthreads();
    if (tid < TOPK) {
        out_idx[row * TOPK + tid] = lidx[tid];
        out_val[row * TOPK + tid] = lval[tid];
    }
}

// ---------------------------------------------------------------------------
// Kernel 2: out[b] = h[b] - softplus(lg) * Σ_j s_j (h·V_j) V_j over top-64 j.
// 256 threads (8 waves) per batch row. Dots via V_WMMA_F32_16X16X4_F32:
//   A = 16 gathered V rows (LDS, padded stride), B = h[k..k+3] broadcast to
//   all 16 columns -> every D column holds the 16 dot partials.
// ---------------------------------------------------------------------------
__global__ __launch_bounds__(NTHR)
void apply_kernel(const float* __restrict__ h,
                  const float* __restrict__ V,
                  const float* __restrict__ log_gamma,
                  const int*   __restrict__ sel_idx,
                  const float* __restrict__ sel_val,
                  float* __restrict__ out)
{
    __shared__ float h_s[HIDDEN];
    __shared__ float Vg[16 * VSTRIDE];
    __shared__ float pp[8][16];
    __shared__ float c_s[16];
    __shared__ int   idx_s[TOPK];
    __shared__ float val_s[TOPK];

    const int tid  = threadIdx.x;
    const int row  = blockIdx.x;
    const int lane = tid & 31;
    const int wave = tid >> 5;

    float hloc[3];
    #pragma unroll
    for (int j = 0; j < 3; ++j) {
        hloc[j] = h[(size_t)row * HIDDEN + tid * 3 + j];
        h_s[tid * 3 + j] = hloc[j];
    }
    if (tid < TOPK) {
        idx_s[tid] = sel_idx[row * TOPK + tid];
        val_s[tid] = sel_val[row * TOPK + tid];
    }
    __syncthreads();

    float raw0 = 0.f, raw1 = 0.f, raw2 = 0.f;

    for (int chunk = 0; chunk < TOPK / 16; ++chunk) {
        // Gather 16 V rows into LDS (L2-resident: V is 96MB < 192MB L2)
        #pragma unroll
        for (int i = 0; i < 12; ++i) {
            int f4   = i * NTHR + tid;          // 0..3071 float4 slots
            int r    = f4 / (HIDDEN / 4);       // /192
            int c4   = f4 % (HIDDEN / 4);
            int vrow = idx_s[chunk * 16 + r];
            float4 v = *(const float4*)(V + (size_t)vrow * HIDDEN + c4 * 4);
            float* dst = &Vg[r * VSTRIDE + c4 * 4];
            dst[0] = v.x; dst[1] = v.y; dst[2] = v.z; dst[3] = v.w;
        }
        __syncthreads();

        // Each wave covers K-slice [wave*96, wave*96+96) in steps of 4.
        v8f acc = {};
        const int m  = lane & 15;
        const int hi = lane >> 4;
        const float* arow = &Vg[m * VSTRIDE];
        #pragma unroll
        for (int it = 0; it < 24; ++it) {
            int k = wave * 96 + it * 4 + 2 * hi;   // per-lane K pair (A/B VGPR layout)
            v2f a = *(const v2f*)(arow + k);       // A: row m of gathered V
            v2f b = *(const v2f*)(&h_s[k]);        // B: h broadcast across N
            acc = __builtin_amdgcn_wmma_f32_16x16x4_f32(
                      false, a, false, b, (short)0, acc, false, false);
        }
        // Column 0 of D = the 16 dot partials (lane 0: M=0..7, lane 16: M=8..15)
        if (lane == 0) {
            #pragma unroll
            for (int r = 0; r < 8; ++r) pp[wave][r] = acc[r];
        } else if (lane == 16) {
            #pragma unroll
            for (int r = 0; r < 8; ++r) pp[wave][8 + r] = acc[r];
        }
        __syncthreads();
        if (tid < 16) {
            float p = 0.f;
            #pragma unroll
            for (int w = 0; w < 8; ++w) p += pp[w][tid];
            c_s[tid] = val_s[chunk * 16 + tid] * p;   // s_j * (h·V_j)
        }
        __syncthreads();

        // raw += c_j * V_j (3 hidden elems per thread, stride-3 -> conflict free)
        #pragma unroll
        for (int j = 0; j < 16; ++j) {
            float cj = c_s[j];
            const float* vr = &Vg[j * VSTRIDE + tid * 3];
            raw0 += cj * vr[0];
            raw1 += cj * vr[1];
            raw2 += cj * vr[2];
        }
        __syncthreads();   // Vg reused next chunk
    }

    float gamma = log1pf(expf(log_gamma[0]));   // softplus
    size_t o = (size_t)row * HIDDEN + tid * 3;
    out[o + 0] = hloc[0] - gamma * raw0;
    out[o + 1] = hloc[1] - gamma * raw1;
    out[o + 2] = hloc[2] - gamma * raw2;
}

// ---------------------------------------------------------------------------
extern "C" void kernel_launch(void* const* d_in, const int* in_sizes, int n_in,
                              void* d_out, int out_size, void* d_ws, size_t ws_size,
                              hipStream_t stream)
{
    (void)in_sizes; (void)n_in; (void)out_size; (void)ws_size;
    const float* h  = (const float*)d_in[0];
    const float* s  = (const float*)d_in[1];
    const float* V  = (const float*)d_in[2];
    const float* lg = (const float*)d_in[3];
    // d_in[4] = top_k (64), baked into the kernel structure.
    float* out = (float*)d_out;

    int*   ws_idx = (int*)d_ws;
    float* ws_val = (float*)((char*)d_ws + (size_t)BATCH * TOPK * sizeof(int));

    size_t shmem = (size_t)NSEEN * 4   // srow
                 + 2048 * 4            // hist
                 + 256 * 4             // bsum
                 + 64 * 4 + 64 * 4     // lidx/lval
                 + 16;                 // vars

    topk_kernel<<<dim3(BATCH), dim3(NTHR), shmem, stream>>>(s, ws_idx, ws_val);
    apply_kernel<<<dim3(BATCH), dim3(NTHR), 0, stream>>>(h, V, lg, ws_idx, ws_val, out);
}